// DKNN_30666066493933
// MI455X (gfx1250) — compile-verified
//
#include <hip/hip_runtime.h>
#include <math.h>

typedef __attribute__((ext_vector_type(2))) float v2f;
typedef __attribute__((ext_vector_type(8))) float v8f;

namespace {
constexpr int QN   = 128;   // queries
constexpr int MN   = 1024;  // neighbors
constexpr int DN   = 256;   // feature dim
constexpr int KTOP = 10;    // top-k rows of P_hat
constexpr float TAU = 1.0f;
constexpr int TILES_M = QN / 16;   // 8
constexpr int TILES_N = MN / 16;   // 64
constexpr int WAVES_PER_BLOCK = 8; // 256 threads
}

// ---------------------------------------------------------------------------
// Kernel 1: scores[q,m] = 2*(Q @ N^T) - |q|^2 - |n|^2 via V_WMMA_F32_16X16X4_F32.
// One wave32 per 16x16 output tile; K-loop of 64 WMMA steps over D=256.
// Row/col L2 norms accumulated inline and folded in the epilogue via shuffles.
// ---------------------------------------------------------------------------
__global__ __launch_bounds__(256) void dknn_scores_wmma(
    const float* __restrict__ query,
    const float* __restrict__ neigh,
    float* __restrict__ scores)
{
    const int lane = threadIdx.x & 31;
    const int wave = threadIdx.x >> 5;
    const int tile = blockIdx.x * WAVES_PER_BLOCK + wave;   // 0..511
    const int tm = tile >> 6;   // tile row (0..7)
    const int tn = tile & 63;   // tile col (0..63)

    const int rsel = lane & 15;          // A row / B col this lane feeds
    const int koff = (lane >> 4) << 1;   // lanes 16-31 take K+2,K+3

    // v2f-aligned row pointers (8B aligned: koff is even)
    const v2f* __restrict__ qv =
        (const v2f*)(query + (size_t)(tm * 16 + rsel) * DN + koff);
    const v2f* __restrict__ nv =
        (const v2f*)(neigh + (size_t)(tn * 16 + rsel) * DN + koff);

    v8f acc = {0.f, 0.f, 0.f, 0.f, 0.f, 0.f, 0.f, 0.f};
    float qp = 0.f;   // partial |q|^2 (half the columns; partner lane has rest)
    float np = 0.f;   // partial |n|^2

#pragma unroll 8
    for (int kk = 0; kk < DN / 4; ++kk) {
        v2f a = qv[2 * kk];   // query[row][4*kk + koff .. +1]
        v2f b = nv[2 * kk];   // neigh[col][4*kk + koff .. +1]
        qp = fmaf(a.x, a.x, fmaf(a.y, a.y, qp));
        np = fmaf(b.x, b.x, fmaf(b.y, b.y, np));
        // D = A(16x4) * B(4x16) + C ; emits v_wmma_f32_16x16x4_f32
        acc = __builtin_amdgcn_wmma_f32_16x16x4_f32(
            /*neg_a=*/false, a, /*neg_b=*/false, b,
            /*c_mod=*/(short)0, acc, /*reuse_a=*/false, /*reuse_b=*/false);
    }

    // Complete the norms: partner half-wave holds the complementary K columns.
    qp += __shfl_xor(qp, 16, 32);   // lane L now holds |query[tm*16 + (L&15)]|^2
    np += __shfl_xor(np, 16, 32);   // lane L now holds |neigh[tn*16 + (L&15)]|^2

    const int halfsel = (lane >> 4) << 3;   // 0 or 8
    float* __restrict__ orow = scores + (size_t)(tm * 16) * MN + tn * 16 + rsel;

#pragma unroll
    for (int v = 0; v < 8; ++v) {
        const int mloc = v + halfsel;               // C/D row for this VGPR
        const float qn2 = __shfl(qp, mloc, 32);     // row norm lives in lane mloc
        const float s = 2.0f * acc[v] - qn2 - np;   // -(squared distance)
        orow[(size_t)mloc * MN] = s;
    }
}

// ---------------------------------------------------------------------------
// Kernel 2: NeuralSort top-K relaxation per query row.
// One block of 1024 threads (32 wave32) per query row; scores row cached in
// LDS (broadcast reads for the O(M^2) pairwise |s_i - s_j| sum), then K=10
// numerically-stabilized softmaxes accumulated and written back over d_out.
// ---------------------------------------------------------------------------
__global__ __launch_bounds__(1024) void dknn_neuralsort(float* __restrict__ out)
{
    __shared__ float sLDS[MN];
    __shared__ float red[32];

    const int q    = blockIdx.x;
    const int tid  = threadIdx.x;
    const int lane = tid & 31;
    const int wid  = tid >> 5;

    const float s = out[(size_t)q * MN + tid];
    sLDS[tid] = s;
    __syncthreads();

    // A_rowsum[m] = sum_j |s[m] - s[j]|  (LDS broadcast, b128 reads)
    float A = 0.f;
    const float4* __restrict__ s4 = (const float4*)sLDS;
#pragma unroll 4
    for (int j = 0; j < MN / 4; ++j) {
        const float4 x = s4[j];
        A += fabsf(s - x.x) + fabsf(s - x.y) + fabsf(s - x.z) + fabsf(s - x.w);
    }

    float accum = 0.f;
#pragma unroll 1
    for (int k = 0; k < KTOP; ++k) {
        const float ck = (float)(MN - 1) - 2.0f * (float)k;
        const float p  = (s * ck - A) * (1.0f / TAU);

        // --- block max ---
        float m = p;
#pragma unroll
        for (int o = 16; o > 0; o >>= 1) m = fmaxf(m, __shfl_xor(m, o, 32));
        if (lane == 0) red[wid] = m;
        __syncthreads();
        if (wid == 0) {
            float x = red[lane];
#pragma unroll
            for (int o = 16; o > 0; o >>= 1) x = fmaxf(x, __shfl_xor(x, o, 32));
            if (lane == 0) red[0] = x;
        }
        __syncthreads();
        m = red[0];
        __syncthreads();

        // --- exp + block sum ---
        const float e = __expf(p - m);
        float t = e;
#pragma unroll
        for (int o = 16; o > 0; o >>= 1) t += __shfl_xor(t, o, 32);
        if (lane == 0) red[wid] = t;
        __syncthreads();
        if (wid == 0) {
            float x = red[lane];
#pragma unroll
            for (int o = 16; o > 0; o >>= 1) x += __shfl_xor(x, o, 32);
            if (lane == 0) red[0] = x;
        }
        __syncthreads();
        const float sum = red[0];
        __syncthreads();

        accum += e / sum;
    }

    out[(size_t)q * MN + tid] = accum;
}

// ---------------------------------------------------------------------------
extern "C" void kernel_launch(void* const* d_in, const int* in_sizes, int n_in,
                              void* d_out, int out_size, void* d_ws, size_t ws_size,
                              hipStream_t stream)
{
    (void)in_sizes; (void)n_in; (void)d_ws; (void)ws_size; (void)out_size;

    const float* query = (const float*)d_in[0];   // [128, 256]
    const float* neigh = (const float*)d_in[1];   // [1024, 256]
    float* out = (float*)d_out;                   // [128, 1024]

    // Phase 1: 512 tiles / 8 waves per block -> 64 blocks. Scores land in d_out.
    dknn_scores_wmma<<<(TILES_M * TILES_N) / WAVES_PER_BLOCK, 32 * WAVES_PER_BLOCK,
                       0, stream>>>(query, neigh, out);

    // Phase 2: one WGP-filling block per query row; overwrites d_out in place.
    dknn_neuralsort<<<QN, MN, 0, stream>>>(out);
}